// QLSTMGen_65481071402563
// MI455X (gfx1250) — compile-verified
//
#include <hip/hip_runtime.h>
#include <math.h>

typedef __attribute__((ext_vector_type(2))) float v2f;
typedef __attribute__((ext_vector_type(4))) float v4f;
typedef __attribute__((ext_vector_type(8))) float v8f;
typedef __attribute__((ext_vector_type(4))) int   v4i;

#define NQ   8
#define QDIM 256
#define Tn   512
#define Bn   128
#define Dn   256
#define Hn   256

// ---------------------------------------------------------------------------
// gfx1250 async global->LDS copy (ASYNCcnt-tracked), with portable fallback.
// Builtin prototype (from hipcc diagnostic): param0 = v4i __device__* (AS1),
// param1 = LDS-side pointer, then imm offset + imm cpol.
// ---------------------------------------------------------------------------
#if __has_builtin(__builtin_amdgcn_global_load_async_to_lds_b128)
#define ASYNC_LDS 1
#else
#define ASYNC_LDS 0
#endif

__device__ __forceinline__ void async_cp16(const float* g, float* l) {
#if ASYNC_LDS
    __builtin_amdgcn_global_load_async_to_lds_b128(
        (__attribute__((address_space(1))) v4i*)g,
        (__attribute__((address_space(3))) v4i*)l, 0, 0);
#else
    *(v4f*)l = *(const v4f*)g;
#endif
}

__device__ __forceinline__ void async_wait8() {
#if ASYNC_LDS
#if __has_builtin(__builtin_amdgcn_s_wait_asynccnt)
    __builtin_amdgcn_s_wait_asynccnt(8);
#else
    asm volatile("s_wait_asynccnt 0x8" ::: "memory");
#endif
#endif
}

__device__ __forceinline__ void async_wait0() {
#if ASYNC_LDS
#if __has_builtin(__builtin_amdgcn_s_wait_asynccnt)
    __builtin_amdgcn_s_wait_asynccnt(0);
#else
    asm volatile("s_wait_asynccnt 0x0" ::: "memory");
#endif
#endif
}

// ---------------------------------------------------------------------------
// Phase 1 (parallel, WMMA): xproj[r][j] = sum_d X[r*256+d] * Wp[j*512+d]
// r = t*B + b (65536 rows). 128-thread blocks = 4 waves; one 16-row tile per
// wave. K=256 split into 4 slabs of 64; each wave double-buffers its own
// 16x64 A-slab in LDS via async-to-LDS b128 (coalesced 16B/lane), then runs
// 16x V_WMMA_F32_16X16X4_F32 per slab off bank-conflict-free ds_load_b64.
// B (Wp x-part) is pre-transposed into LDS with zero-padded N=8..15 columns.
// ---------------------------------------------------------------------------
#define ASTRIDE 68   // padded row stride (floats): conflict-free frag reads

__global__ __launch_bounds__(128) void xproj_wmma_kernel(
    const float* __restrict__ X, const float* __restrict__ Wp,
    float* __restrict__ xproj)
{
    __shared__ float sh_B2[128 * 32];          // [m=k/2][n*2+e], 16 KB
    __shared__ float sh_A[4][2][16 * ASTRIDE]; // per-wave double buffer

    const int tid  = threadIdx.x;
    const int wave = tid >> 5;
    const int lane = tid & 31;
    const int tile = blockIdx.x * 4 + wave;    // 4096 tiles total
    const int row0 = tile * 16;

    const int r15 = lane & 15;                 // A row in tile / B column
    const int g   = lane >> 4;                 // K-pair group (0 or 1)

    // ---- preload B^T into LDS: sh_B2[m*32 + 2n + e] = Wp[n][2m+e], 0-pad --
    for (int i = tid; i < 128 * 32; i += 128) {
        const int m = i >> 5, n2 = i & 31, n = n2 >> 1, e = n2 & 1;
        sh_B2[i] = (n < NQ) ? Wp[n * 512 + 2 * m + e] : 0.0f;
    }

    // ---- stage slab 0 (wave-private region, no barrier needed for it) ----
    {
        #pragma unroll
        for (int chunk = 0; chunk < 8; ++chunk) {
            const int idx = chunk * 32 + lane;
            const int r = idx >> 4, cq = idx & 15;
            async_cp16(X + (size_t)(row0 + r) * Dn + cq * 4,
                       &sh_A[wave][0][r * ASTRIDE + cq * 4]);
        }
    }
    __syncthreads();                           // B2 visible to all waves

    v8f acc = {};
    for (int ks = 0; ks < 4; ++ks) {
        if (ks < 3) {                          // prefetch next slab -> other buf
            const int kb = (ks + 1) * 64;
            float* dst = &sh_A[wave][(ks + 1) & 1][0];
            #pragma unroll
            for (int chunk = 0; chunk < 8; ++chunk) {
                const int idx = chunk * 32 + lane;
                const int r = idx >> 4, cq = idx & 15;
                async_cp16(X + (size_t)(row0 + r) * Dn + kb + cq * 4,
                           dst + r * ASTRIDE + cq * 4);
            }
            async_wait8();                     // current slab's 8 xfers done
        } else {
            async_wait0();
        }

        const float* Ab = &sh_A[wave][ks & 1][0];
        const int kbase = ks * 64;
        #pragma unroll
        for (int kk = 0; kk < 64; kk += 4) {
            v2f a = *(const v2f*)(Ab + r15 * ASTRIDE + kk + 2 * g);
            v2f b = *(const v2f*)(sh_B2 + (((kbase + kk) >> 1) + g) * 32 + 2 * r15);
            acc = __builtin_amdgcn_wmma_f32_16x16x4_f32(
                false, a, false, b, (short)0, acc, false, false);
        }
    }

    // D layout: VGPR v -> (M=v, N=lane) lanes 0-15, (M=v+8, N=lane-16) hi.
    const int  col = lane & 15;
    const bool hi  = lane >= 16;
    if (col < NQ) {
        #pragma unroll
        for (int v = 0; v < 8; ++v) {
            const int m = v + (hi ? 8 : 0);
            xproj[(size_t)(row0 + m) * NQ + col] = acc[v];
        }
    }
}

// ---------------------------------------------------------------------------
// Phase 2 (sequential scan): one workgroup per batch element b, 256 threads
// (8 wave32), t = 0..511. All recurrent state + weights live in LDS.
// 3 barriers per timestep.
// ---------------------------------------------------------------------------
__global__ __launch_bounds__(256) void qlstm_scan_kernel(
    const float* __restrict__ xproj, const float* __restrict__ hx,
    const float* __restrict__ cx,    const float* __restrict__ Wp,
    const float* __restrict__ qp,    const float* __restrict__ Wg,
    const float* __restrict__ bg,    float* __restrict__ out)
{
    __shared__ float sh_h[Hn];
    __shared__ float sh_c[Hn];
    __shared__ float sh_Wph[NQ][Hn];                 // Wp[:, 256:512] (8 KB)
    __shared__ float sh_WgT[NQ][4 * Hn];             // Wg transposed (32 KB)
    __shared__ float sh_bg[4 * Hn];                  //               (4 KB)
    __shared__ float sh_qp[NQ];
    __shared__ float sh_cs[NQ][2];                   // cos^2 / sin^2 per wire
    __shared__ int   sh_inv[QDIM];                   // inverse composed perm
    __shared__ __align__(16) float sh_wpart[NQ][8];  // [qubit][wave] partials

    const int tid  = threadIdx.x;
    const int b    = blockIdx.x;
    const int wave = tid >> 5;
    const int lane = tid & 31;

    // ---- preload weights into LDS -----------------------------------------
    for (int i = tid; i < NQ * Hn; i += 256)
        sh_Wph[i >> 8][i & 255] = Wp[(i >> 8) * 512 + 256 + (i & 255)];
    for (int i = tid; i < NQ * 4 * Hn; i += 256) {
        const int j = i >> 10, gidx = i & 1023;
        sh_WgT[j][gidx] = Wg[gidx * NQ + j];
    }
    for (int i = tid; i < 4 * Hn; i += 256) sh_bg[i] = bg[i];
    if (tid < NQ) sh_qp[tid] = qp[tid];
    sh_h[tid] = hx[b * Hn + tid];
    sh_c[tid] = cx[b * Hn + tid];

    // ---- composed CNOT-ring permutation: composite(i) = p0(p1(...p7(i))) --
    {
        int m = tid;
        for (int c = NQ - 1; c >= 0; --c) {
            const int tq   = (c + 1) & (NQ - 1);
            const int cbit = (m >> (NQ - 1 - c)) & 1;
            if (cbit) m ^= 1 << (NQ - 1 - tq);
        }
        sh_inv[m] = tid;                             // inverse permutation
    }
    __syncthreads();

    for (int t = 0; t < Tn; ++t) {
        // ---- step A: proj[j] = xproj + h . Wph[j]  (wave j owns dot j) ----
        {
            const int j = wave;
            float part = 0.0f;
            #pragma unroll
            for (int i = 0; i < 8; ++i) {
                const int k = lane + 32 * i;
                part += sh_h[k] * sh_Wph[j][k];
            }
            #pragma unroll
            for (int m = 16; m >= 1; m >>= 1)
                part += __shfl_xor(part, m, 32);
            if (lane == 0) {
                const float proj = part + xproj[((size_t)t * Bn + b) * NQ + j];
                const float th   = 0.5f * (proj + sh_qp[j]);
                float s, c;
                __sincosf(th, &s, &c);
                sh_cs[j][0] = c * c;                 // |a0|^2
                sh_cs[j][1] = s * s;                 // |a1|^2
            }
            if (tid == 0 && t + 1 < Tn)              // -> global_prefetch_b8
                __builtin_prefetch(&xproj[((size_t)(t + 1) * Bn + b) * NQ], 0, 0);
        }
        __syncthreads();

        // ---- step B: probs (product form) + per-wave Z partial sums -------
        {
            const int s = tid;                       // one basis state/thread
            float prob = 1.0f;
            #pragma unroll
            for (int w = 0; w < NQ; ++w)
                prob *= sh_cs[w][(s >> (NQ - 1 - w)) & 1];
            const int ii = sh_inv[s];
            float q[NQ];
            #pragma unroll
            for (int j = 0; j < NQ; ++j)
                q[j] = (ii & (1 << (NQ - 1 - j))) ? -prob : prob;
            #pragma unroll
            for (int m = 16; m >= 1; m >>= 1) {
                #pragma unroll
                for (int j = 0; j < NQ; ++j)
                    q[j] += __shfl_xor(q[j], m, 32);
            }
            if (lane == 0) {
                #pragma unroll
                for (int j = 0; j < NQ; ++j) sh_wpart[j][wave] = q[j];
            }
        }
        __syncthreads();

        // ---- step C: fold partials, gates = qout.WgT + bg, LSTM pointwise -
        {
            float qv[NQ];
            #pragma unroll
            for (int j = 0; j < NQ; ++j) {           // 2x ds_load_b128 each
                v4f p0 = *(const v4f*)&sh_wpart[j][0];
                v4f p1 = *(const v4f*)&sh_wpart[j][4];
                qv[j] = ((p0.x + p0.y) + (p0.z + p0.w)) +
                        ((p1.x + p1.y) + (p1.z + p1.w));
            }

            float gf = sh_bg[tid];
            float gi = sh_bg[Hn + tid];
            float gg = sh_bg[2 * Hn + tid];
            float go = sh_bg[3 * Hn + tid];
            #pragma unroll
            for (int j = 0; j < NQ; ++j) {
                gf += qv[j] * sh_WgT[j][tid];
                gi += qv[j] * sh_WgT[j][Hn + tid];
                gg += qv[j] * sh_WgT[j][2 * Hn + tid];
                go += qv[j] * sh_WgT[j][3 * Hn + tid];
            }
            const float f  = 1.0f / (1.0f + __expf(-gf));
            const float ig = 1.0f / (1.0f + __expf(-gi));
            const float o  = 1.0f / (1.0f + __expf(-go));
            const float cn = f * sh_c[tid] + ig * tanhf(gg);
            const float hn = o * tanhf(cn);
            sh_c[tid] = cn;
            sh_h[tid] = hn;  // safe: all reads of old h ended at barrier A
            out[((size_t)t * Bn + b) * Hn + tid] = hn;
        }
        __syncthreads();
    }

    // ---- finals: h_fin, c_fin ---------------------------------------------
    out[(size_t)Tn * Bn * Hn + (size_t)b * Hn + tid] = sh_h[tid];
    out[(size_t)Tn * Bn * Hn + (size_t)Bn * Hn + (size_t)b * Hn + tid] = sh_c[tid];
}

// ---------------------------------------------------------------------------
extern "C" void kernel_launch(void* const* d_in, const int* in_sizes, int n_in,
                              void* d_out, int out_size, void* d_ws, size_t ws_size,
                              hipStream_t stream)
{
    const float* X  = (const float*)d_in[0];   // (512,128,256)
    const float* hx = (const float*)d_in[1];   // (128,256)
    const float* cx = (const float*)d_in[2];   // (128,256)
    const float* Wp = (const float*)d_in[3];   // (8,512)
    const float* qp = (const float*)d_in[4];   // (8,)
    const float* Wg = (const float*)d_in[5];   // (1024,8)
    const float* bg = (const float*)d_in[6];   // (1024,)
    float* out   = (float*)d_out;
    float* xproj = (float*)d_ws;               // 65536*8 f32 = 2 MB scratch

    // Phase 1: WMMA GEMM, async-LDS double-buffered (HBM-bound: ~3 us)
    xproj_wmma_kernel<<<1024, 128, 0, stream>>>(X, Wp, xproj);
    // Phase 2: 128 independent per-batch recurrences (latency-bound scan)
    qlstm_scan_kernel<<<Bn, 256, 0, stream>>>(xproj, hx, cx, Wp, qp, Wg, bg, out);
}